// CalibrationLoss_14637248544967
// MI455X (gfx1250) — compile-verified
//
#include <hip/hip_runtime.h>

// ECE (expected calibration error) over B = 2^25 f32 elements.
// Phase 1: bandwidth-bound streaming histogram (global_load_b128 + conflict-free
//          lane-private ds_add_f32 histograms), per-block partials -> d_ws.
// Phase 2: single-wave deterministic reduction of the 1024x20 partial table on
//          the matrix pipe via V_WMMA_F32_16X16X4_F32 (A = ones => WMMA acts as
//          a layout-robust 64-wide f32 adder tree; sum of all D = 16 * sum(B)).

#define GRID   1024
#define BLOCK  256     // 8 waves of 32
#define NB     10      // real bins; slot 10 = overflow (dropped)
#define SLOTS  11      // bins incl. overflow
#define WSTRIDE 704    // per-wave LDS floats: 2 arrays * 11 bins * 32 lanes

typedef __attribute__((ext_vector_type(2))) float v2f;
typedef __attribute__((ext_vector_type(8))) float v8f;

__global__ __launch_bounds__(BLOCK)
void ece_hist_kernel(const float* __restrict__ conf,
                     const float* __restrict__ corr,
                     float* __restrict__ partial,
                     unsigned nchunk) {
  __shared__ float hist[8 * WSTRIDE];   // [wave][ conf 11x32 | corr 11x32 ]
  __shared__ float red[20 * 8];

  const int lane  = threadIdx.x & 31;
  const int wave  = threadIdx.x >> 5;
  const int wbase = wave * WSTRIDE;

  // Zero this lane's private slots (lane-private -> in-order per wave).
  #pragma unroll
  for (int b = 0; b < SLOTS; ++b) {
    hist[wbase + b * 32 + lane]       = 0.0f;
    hist[wbase + 352 + b * 32 + lane] = 0.0f;
  }
  __syncthreads();

  const float4* __restrict__ c4 = (const float4*)conf;
  const float4* __restrict__ r4 = (const float4*)corr;
  const unsigned stride = gridDim.x * blockDim.x;   // 262144
  unsigned t = blockIdx.x * blockDim.x + threadIdx.x;

  #pragma unroll 4
  for (unsigned i = t; i < nchunk; i += stride) {
    float4 c = c4[i];
    float4 r = r4[i];
    float cv[4] = {c.x, c.y, c.z, c.w};
    float rv[4] = {r.x, r.y, r.z, r.w};
    #pragma unroll
    for (int j = 0; j < 4; ++j) {
      // idx = floor(c*10) (c >= 0 so f32->u32 trunc == floor), clamp to 10.
      unsigned idx = (unsigned)(cv[j] * 10.0f);
      idx = idx > 10u ? 10u : idx;
      // bank = (idx*32 + lane) % 64 -> conflict-free across lanes of a wave.
      int off = wbase + (int)idx * 32 + lane;
      atomicAdd(&hist[off], cv[j]);         // ds_add_f32, lane-private slot
      atomicAdd(&hist[off + 352], rv[j]);   // ds_add_f32, lane-private slot
    }
  }
  __syncthreads();

  // Deterministic in-block reduction: 20 logical bins (10 conf + 10 corr).
  int tid = threadIdx.x;
  if (tid < 160) {
    int b = tid >> 3;       // 0..19
    int w = tid & 7;        // wave
    int off = w * WSTRIDE + (b < NB ? b * 32 : 352 + (b - NB) * 32);
    float s = 0.0f;
    #pragma unroll
    for (int l = 0; l < 32; ++l) s += hist[off + l];
    red[b * 8 + w] = s;
  }
  __syncthreads();
  if (tid < 20) {
    float s = 0.0f;
    #pragma unroll
    for (int w = 0; w < 8; ++w) s += red[tid * 8 + w];
    partial[blockIdx.x * 20 + tid] = s;     // fixed slot -> deterministic
  }
}

// One wave32; EXEC all-1s and uniform control flow around the WMMAs.
__global__ __launch_bounds__(32)
void ece_final_kernel(const float* __restrict__ partial,
                      float* __restrict__ out,
                      float n) {
  __shared__ float sArr[20 * 32];
  const int lane = threadIdx.x;

  v2f ones;
  ones.x = 1.0f; ones.y = 1.0f;               // A = all ones (layout-irrelevant)

  for (int b = 0; b < 20; ++b) {              // uniform loop
    v8f acc = {};
    for (int g = 0; g < 16; ++g) {            // 16 chunks * 64 values = 1024
      v2f bv;
      bv.x = partial[(unsigned)(g * 64 + lane)        * 20u + (unsigned)b];
      bv.y = partial[(unsigned)(g * 64 + 32 + lane)   * 20u + (unsigned)b];
      // D = A(ones) x B + C : every D element = a column-sum of B; the sum of
      // all 256 D elements = 16 * (sum of all 64 B values), layout-robust.
      acc = __builtin_amdgcn_wmma_f32_16x16x4_f32(
          /*neg_a=*/false, ones, /*neg_b=*/false, bv,
          /*c_mod=*/(short)0, acc, /*reuse_a=*/false, /*reuse_b=*/false);
    }
    float s = acc[0] + acc[1] + acc[2] + acc[3] +
              acc[4] + acc[5] + acc[6] + acc[7];
    sArr[b * 32 + lane] = s;
  }
  __syncthreads();

  if (lane == 0) {
    float e = 0.0f;
    for (int b = 0; b < NB; ++b) {
      float tc = 0.0f, tr = 0.0f;
      #pragma unroll
      for (int l = 0; l < 32; ++l) {
        tc += sArr[b * 32 + l];
        tr += sArr[(b + NB) * 32 + l];
      }
      e += fabsf(tc - tr);                    // 16*|Sconf - Scorr|
    }
    out[0] = e / 16.0f / n;                   // /16 undoes WMMA replication
  }
}

extern "C" void kernel_launch(void* const* d_in, const int* in_sizes, int n_in,
                              void* d_out, int out_size, void* d_ws, size_t ws_size,
                              hipStream_t stream) {
  const float* conf = (const float*)d_in[0];
  const float* corr = (const float*)d_in[1];
  float* out        = (float*)d_out;
  float* partial    = (float*)d_ws;           // GRID*20 floats = 80 KB
  int n = in_sizes[0];                        // 33554432

  ece_hist_kernel<<<GRID, BLOCK, 0, stream>>>(conf, corr, partial,
                                              (unsigned)(n >> 2));
  ece_final_kernel<<<1, 32, 0, stream>>>(partial, out, (float)n);
}